// LearnableGate_46789373723355
// MI455X (gfx1250) — compile-verified
//
#include <hip/hip_runtime.h>
#include <math.h>

// Problem constants (match reference)
#define N_FEAT  4096
#define K_TOP   256
#define OUT_CH  512
#define BATCH   16
#define BLK     256
#define VPT     (N_FEAT / BLK)   // 16 values per thread in kernel 1
#define NWAVES  (BLK / 32)       // 8 waves (wave32)

typedef float v4f __attribute__((ext_vector_type(4)));

#if __has_builtin(__builtin_amdgcn_global_load_async_to_lds_b32)
#define HAVE_ASYNC_LDS 1
typedef __attribute__((address_space(1))) int* gptr_i32;
typedef __attribute__((address_space(3))) int* lptr_i32;
#else
#define HAVE_ASYNC_LDS 0
#endif

// ---------------- block reductions (wave32) ----------------

__device__ __forceinline__ float blk_sum_f(float v, volatile float* sh) {
#pragma unroll
  for (int off = 16; off; off >>= 1) v += __shfl_down(v, off, 32);
  __syncthreads();
  if ((threadIdx.x & 31) == 0) sh[threadIdx.x >> 5] = v;
  __syncthreads();
  float tot = 0.f;
#pragma unroll
  for (int i = 0; i < NWAVES; ++i) tot += sh[i];
  return tot;
}

__device__ __forceinline__ float blk_max_f(float v, volatile float* sh) {
#pragma unroll
  for (int off = 16; off; off >>= 1) v = fmaxf(v, __shfl_down(v, off, 32));
  __syncthreads();
  if ((threadIdx.x & 31) == 0) sh[threadIdx.x >> 5] = v;
  __syncthreads();
  float tot = sh[0];
#pragma unroll
  for (int i = 1; i < NWAVES; ++i) tot = fmaxf(tot, sh[i]);
  return tot;
}

__device__ __forceinline__ int blk_sum_i(int v, volatile int* sh) {
#pragma unroll
  for (int off = 16; off; off >>= 1) v += __shfl_down(v, off, 32);
  __syncthreads();
  if ((threadIdx.x & 31) == 0) sh[threadIdx.x >> 5] = v;
  __syncthreads();
  int tot = 0;
#pragma unroll
  for (int i = 0; i < NWAVES; ++i) tot += sh[i];
  return tot;
}

// ---------------- kernel 1: per-column stats ----------------
// One block per output channel o. scores layout: [n, o] row-major (OUT_CH fastest).
// Produces ws[o]          = column max m
//          ws[512 + o]    = 1/Z   (Z = sum_n exp(2*(s-m)))
//          ws[1024 + o]   = bitcast(threshold bits)  (K-th largest score)
//          ws[1536 + o]   = 1/S   (S = sum over kept of ((soft - s) + s))

__global__ void __launch_bounds__(BLK)
gate_colstats(const float* __restrict__ scores, float* __restrict__ ws) {
  const int o = blockIdx.x;
  const int t = threadIdx.x;

  __shared__ float lds_vals[N_FEAT];   // 16 KB column staging
  __shared__ float fsh[NWAVES];
  __shared__ int   ish[NWAVES];

  float v[VPT];

#if HAVE_ASYNC_LDS
  // CDNA5 async global->LDS gather of the strided column, no VGPR staging.
#pragma unroll
  for (int i = 0; i < VPT; ++i) {
    const int n = t + i * BLK;
    __builtin_amdgcn_global_load_async_to_lds_b32(
        (gptr_i32)(scores + (size_t)n * OUT_CH + o),
        (lptr_i32)(&lds_vals[n]),
        0, 0);
  }
#if __has_builtin(__builtin_amdgcn_s_wait_asynccnt)
  __builtin_amdgcn_s_wait_asynccnt(0);
#else
  asm volatile("s_wait_asynccnt 0" ::: "memory");
#endif
  __syncthreads();
#pragma unroll
  for (int i = 0; i < VPT; ++i) v[i] = lds_vals[t + i * BLK];
#else
  (void)lds_vals;
#pragma unroll
  for (int i = 0; i < VPT; ++i) v[i] = scores[(size_t)(t + i * BLK) * OUT_CH + o];
#endif

  // 1) column max
  float lm = -1e30f;
#pragma unroll
  for (int i = 0; i < VPT; ++i) lm = fmaxf(lm, v[i]);
  const float m = blk_max_f(lm, fsh);

  // 2) softmax partition function (T = 0.5 -> multiply exponent by 2)
  float e[VPT];
  float ls = 0.f;
#pragma unroll
  for (int i = 0; i < VPT; ++i) {
    e[i] = __expf((v[i] - m) * 2.0f);
    ls += e[i];
  }
  const float Z  = blk_sum_f(ls, fsh);
  const float rZ = 1.0f / Z;

  // 3) K-th largest score: radix bit-search on IEEE bits (all scores in (0,1) => positive,
  //    so uint bit pattern ordering == float ordering). Find largest tb with count(>=tb) >= K.
  unsigned ub[VPT];
#pragma unroll
  for (int i = 0; i < VPT; ++i) ub[i] = __float_as_uint(v[i]);

  unsigned tb = 0u;
  for (int b = 30; b >= 0; --b) {
    const unsigned cand = tb | (1u << b);
    int c = 0;
#pragma unroll
    for (int i = 0; i < VPT; ++i) c += (ub[i] >= cand) ? 1 : 0;
    const int cnt = blk_sum_i(c, ish);
    if (cnt >= K_TOP) tb = cand;
  }

  // 4) normalization sum over kept positions, mirroring the STE arithmetic
  float lg = 0.f;
#pragma unroll
  for (int i = 0; i < VPT; ++i) {
    if (ub[i] >= tb) {
      const float soft = e[i] * rZ;
      lg += (soft - v[i]) + v[i];
    }
  }
  const float S = blk_sum_f(lg, fsh);

  if (t == 0) {
    ws[o]              = m;
    ws[OUT_CH + o]     = rZ;
    ws[2 * OUT_CH + o] = __uint_as_float(tb);
    ws[3 * OUT_CH + o] = 1.0f / S;
  }
}

// ---------------- kernel 2: compute once, broadcast 16 batches ----------------

__device__ __forceinline__ float gate1(float s, float m, float rZ, float thf, float rS) {
  const float soft = __expf((s - m) * 2.0f) * rZ;
  const float g = (soft - s) + s;            // STE: exact 0 when dropped, ~soft when kept
  return (__float_as_uint(s) >= __float_as_uint(thf)) ? g * rS : 0.0f;
}

__global__ void __launch_bounds__(BLK)
gate_broadcast(const float* __restrict__ scores, const float* __restrict__ ws,
               float* __restrict__ out) {
  const int idx = blockIdx.x * BLK + threadIdx.x;   // v4f index into (N, OUT) plane
  const int c4  = idx & (OUT_CH / 4 - 1);           // v4f column group

  const v4f s4 = reinterpret_cast<const v4f*>(scores)[idx];
  const v4f m4 = reinterpret_cast<const v4f*>(ws)[c4];
  const v4f z4 = reinterpret_cast<const v4f*>(ws + OUT_CH)[c4];
  const v4f t4 = reinterpret_cast<const v4f*>(ws + 2 * OUT_CH)[c4];
  const v4f r4 = reinterpret_cast<const v4f*>(ws + 3 * OUT_CH)[c4];

  v4f g;
  g.x = gate1(s4.x, m4.x, z4.x, t4.x, r4.x);
  g.y = gate1(s4.y, m4.y, z4.y, t4.y, r4.y);
  g.z = gate1(s4.z, m4.z, z4.z, t4.z, r4.z);
  g.w = gate1(s4.w, m4.w, z4.w, t4.w, r4.w);

  v4f* op = reinterpret_cast<v4f*>(out) + idx;
#pragma unroll
  for (int b = 0; b < BATCH; ++b) {
    __builtin_nontemporal_store(g, op);             // NT b128 store: pure streaming output
    op += (N_FEAT * OUT_CH / 4);
  }
}

// ---------------- launch ----------------

extern "C" void kernel_launch(void* const* d_in, const int* in_sizes, int n_in,
                              void* d_out, int out_size, void* d_ws, size_t ws_size,
                              hipStream_t stream) {
  (void)in_sizes; (void)n_in; (void)out_size; (void)ws_size;
  // d_in[0] = X (B,N,8,1,1,1) float32 -- data unused by the reference math
  const float* scores = (const float*)d_in[1];      // (N, OUT) float32
  float* out = (float*)d_out;                       // (B, N, OUT) float32
  float* ws  = (float*)d_ws;                        // 4*OUT_CH floats of column params

  gate_colstats<<<OUT_CH, BLK, 0, stream>>>(scores, ws);

  const int nblk = (N_FEAT * OUT_CH / 4) / BLK;     // 2048
  gate_broadcast<<<nblk, BLK, 0, stream>>>(scores, ws, out);
}